// NeighborPooling_24704651886665
// MI455X (gfx1250) — compile-verified
//
#include <hip/hip_runtime.h>

// NeighborPooling for MI455X (gfx1250, wave32, WMMA).
// Dominant work: e @ W1[128:256]  ==  [B*S*K,128] x [128,128] fp32 GEMM,
// done with V_WMMA_F32_16X16X4_F32, one 16x128x128 tile per wave (node).

#define ALPHA 0.2f
#define B_ 1024
#define S_ 50
#define H_ 128
#define K_ 12
#define NODES (B_ * S_)      // 51200
#define WAVES 8              // waves (nodes) per workgroup
#define WSTR 132             // LDS stride for W1midT rows (conflict-free, 16B-aligned)
#define ESTR 132             // LDS stride for e-tile rows

typedef float v2f __attribute__((ext_vector_type(2)));
typedef float v8f __attribute__((ext_vector_type(8)));

// ---- kernel 1: offsets[b] = first node index of session b (batch is sorted) ----
__global__ void k_offsets(const int* __restrict__ batch, int* __restrict__ offsets) {
    int i = blockIdx.x * blockDim.x + threadIdx.x;
    if (i >= NODES) return;
    int b = batch[i];
    if (i == 0 || batch[i - 1] != b) offsets[b] = i;
}

// ---- kernel 2: masked session mean [B,H] ----
__global__ void k_mean(const float* __restrict__ x_embed, const float* __restrict__ mask,
                       const int* __restrict__ alias_s, const int* __restrict__ offsets,
                       float* __restrict__ smean) {
    int b = blockIdx.x, h = threadIdx.x;
    int off = offsets[b];
    float acc = 0.f, msum = 0.f;
    for (int s = 0; s < S_; ++s) {
        float mk = mask[b * S_ + s];
        int row = off + alias_s[b * S_ + s];
        acc += x_embed[(size_t)row * H_ + h] * mk;
        msum += mk;
    }
    smean[b * H_ + h] = acc / msum;
}

// ---- kernel 3: mcontrib[b,:] = session_mean[b,:] @ W1[0:H,:] + b1 ----
__global__ void k_mcontrib(const float* __restrict__ smean, const float* __restrict__ w1,
                           const float* __restrict__ b1, float* __restrict__ mc) {
    __shared__ float sm[H_];
    int b = blockIdx.x, c = threadIdx.x;
    sm[c] = smean[b * H_ + c];
    __syncthreads();
    float acc = b1[c];
    for (int h = 0; h < H_; ++h) acc += sm[h] * w1[h * H_ + c];
    mc[b * H_ + c] = acc;
}

// ---- kernel 4: main fused kernel. One wave == one node (b,s). ----
__global__ __launch_bounds__(256, 1) void k_main(
        const float* __restrict__ x_embed, const float* __restrict__ mask,
        const float* __restrict__ weight,  const float* __restrict__ emb,
        const float* __restrict__ w1,      const float* __restrict__ w2,
        const int*   __restrict__ session, const int* __restrict__ alias_s,
        const int*   __restrict__ adj,     const int* __restrict__ offsets,
        const float* __restrict__ mc,
        float* __restrict__ gout, float* __restrict__ lout) {

    // W1midT[col][k] = w1[(H+k)*H + col] so B-fragment (k,k+1) pair is contiguous.
    __shared__ __align__(16) float sW[H_ * WSTR];            // 66 KB
    __shared__ __align__(16) float sE[WAVES * 16 * ESTR];    // 66 KB (8 x 16x128 tiles)
    __shared__ float sWgt[WAVES * 16];

    const int tid  = threadIdx.x;
    const int lane = tid & 31, wave = tid >> 5;
    const int half = lane >> 4, l15 = lane & 15;

    // Cooperative load of W1mid (transposed) into LDS, coalesced on global reads.
    for (int idx = tid; idx < H_ * H_; idx += 256) {
        int k = idx >> 7, c = idx & 127;
        sW[c * WSTR + k] = w1[(H_ + k) * H_ + c];
    }

    const int g    = blockIdx.x * WAVES + wave;   // node id in [0, B*S)
    const int item = session[g];

    if (lane < 16) sWgt[wave * 16 + lane] = (lane < K_) ? weight[item * K_ + lane] : 0.f;

    // Gather 12 neighbor embedding rows (zero-pad to 16) into this wave's LDS tile.
    float* eBase = sE + wave * (16 * ESTR);
    #pragma unroll
    for (int k = 0; k < K_; ++k) {
        int row = adj[item * K_ + k];
        const float4 v = *(const float4*)(emb + (size_t)row * H_ + lane * 4);
        *(float4*)(eBase + k * ESTR + lane * 4) = v;
    }
    const float4 z4 = {0.f, 0.f, 0.f, 0.f};
    #pragma unroll
    for (int k = K_; k < 16; ++k)
        *(float4*)(eBase + k * ESTR + lane * 4) = z4;

    __syncthreads();

    // ---- 16x128x128 fp32 GEMM via V_WMMA_F32_16X16X4_F32 ----
    // A 16x4: v0 = A[row][k0+2h], v1 = A[row][k0+2h+1], row = lane&15, h = lane>>4.
    // B 4x16: v0 = B[k0+2h][col], v1 = B[k0+2h+1][col], col = 16f + (lane&15).
    const v8f zacc = {0.f, 0.f, 0.f, 0.f, 0.f, 0.f, 0.f, 0.f};
    v8f acc[8];
    #pragma unroll
    for (int f = 0; f < 8; ++f) acc[f] = zacc;

    const float* aPtr = eBase + l15 * ESTR + 2 * half;
    #pragma unroll 4
    for (int k0 = 0; k0 < H_; k0 += 4) {
        const v2f a = *(const v2f*)(aPtr + k0);
        #pragma unroll
        for (int f = 0; f < 8; ++f) {
            const v2f bf = *(const v2f*)(sW + (16 * f + l15) * WSTR + k0 + 2 * half);
            acc[f] = __builtin_amdgcn_wmma_f32_16x16x4_f32(
                false, a, false, bf, (short)0, acc[f], false, false);
        }
    }

    // ---- epilogue: score per row, leaky-relu, dot w2, softmax over K ----
    const int b = g / S_, s = g - b * S_;

    float wk[8];
    #pragma unroll
    for (int r = 0; r < 8; ++r) wk[r] = sWgt[wave * 16 + r + 8 * half];

    float partial[8];
    #pragma unroll
    for (int r = 0; r < 8; ++r) partial[r] = 0.f;

    #pragma unroll
    for (int f = 0; f < 8; ++f) {
        const int col   = 16 * f + l15;
        const float mcv = mc[b * H_ + col];            // mean@W1_top + b1
        const float wl  = w1[2 * H_ * H_ + col];       // W1 last row (weight column)
        const float w2v = w2[col];
        #pragma unroll
        for (int r = 0; r < 8; ++r) {
            float t = acc[f][r] + mcv + wk[r] * wl;
            t = (t > 0.f) ? t : ALPHA * t;             // leaky_relu
            partial[r] += t * w2v;
        }
    }

    // Reduce over the 16 lanes of each half (rows 0-7 live in lanes 0-15, 8-15 in 16-31).
    #pragma unroll
    for (int r = 0; r < 8; ++r)
        for (int m = 1; m < 16; m <<= 1)
            partial[r] += __shfl_xor(partial[r], m, 32);

    // Broadcast the 12 real scores to all lanes.
    float sc[K_];
    #pragma unroll
    for (int k = 0; k < K_; ++k)
        sc[k] = __shfl(partial[k & 7], (k >> 3) << 4, 32);

    float mx = sc[0];
    #pragma unroll
    for (int k = 1; k < K_; ++k) mx = fmaxf(mx, sc[k]);
    float sum = 0.f;
    #pragma unroll
    for (int k = 0; k < K_; ++k) { sc[k] = __expf(sc[k] - mx); sum += sc[k]; }
    const float inv = 1.f / sum;

    // neighbor_vector + session_embed -> output (each lane owns 4 h-columns).
    const int h0 = lane * 4;
    float4 nv = {0.f, 0.f, 0.f, 0.f};
    #pragma unroll
    for (int k = 0; k < K_; ++k) {
        const float4 e4 = *(const float4*)(eBase + k * ESTR + h0);
        const float a_ = sc[k] * inv;
        nv.x += a_ * e4.x; nv.y += a_ * e4.y; nv.z += a_ * e4.z; nv.w += a_ * e4.w;
    }

    const int   off = offsets[b];
    const int   al  = alias_s[g];
    const float mk  = mask[g];
    const float4 xe = *(const float4*)(x_embed + (size_t)(off + al) * H_ + h0);
    float4 o;
    o.x = xe.x * mk + nv.x; o.y = xe.y * mk + nv.y;
    o.z = xe.z * mk + nv.z; o.w = xe.w * mk + nv.w;
    *(float4*)(gout + (size_t)g * H_ + h0) = o;
    if (s == 0) *(float4*)(lout + (size_t)b * H_ + h0) = o;   // last_item_pre
}

extern "C" void kernel_launch(void* const* d_in, const int* in_sizes, int n_in,
                              void* d_out, int out_size, void* d_ws, size_t ws_size,
                              hipStream_t stream) {
    const float* x_embed = (const float*)d_in[0];
    const float* mask    = (const float*)d_in[1];
    const float* weight  = (const float*)d_in[2];
    const float* emb     = (const float*)d_in[3];
    const float* w1      = (const float*)d_in[4];
    const float* b1      = (const float*)d_in[5];
    const float* w2      = (const float*)d_in[6];
    const int*   session = (const int*)d_in[7];
    const int*   alias_s = (const int*)d_in[8];
    const int*   batch   = (const int*)d_in[9];
    const int*   adj     = (const int*)d_in[10];

    int*   offsets  = (int*)d_ws;                           // B ints
    float* smean    = (float*)((char*)d_ws + 4096);         // B*H floats
    float* mcontrib = smean + (size_t)B_ * H_;              // B*H floats

    float* gout = (float*)d_out;                            // [B*S, H]
    float* lout = gout + (size_t)NODES * H_;                // [B, H]

    k_offsets <<<(NODES + 255) / 256, 256, 0, stream>>>(batch, offsets);
    k_mean    <<<B_, H_, 0, stream>>>(x_embed, mask, alias_s, offsets, smean);
    k_mcontrib<<<B_, H_, 0, stream>>>(smean, w1, b1, mcontrib);
    k_main    <<<NODES / WAVES, 256, 0, stream>>>(x_embed, mask, weight, emb, w1, w2,
                                                  session, alias_s, adj, offsets,
                                                  mcontrib, gout, lout);
}